// RandomAffine_10651518894838
// MI455X (gfx1250) — compile-verified
//
#include <hip/hip_runtime.h>
#include <math.h>
#include <stdint.h>

// 4x sequential bilinear affine warps (rot, scale, shear_x, shear_y) on
// (32,3,512,512) fp32, rolls (8,12) fused into the final pass's store.
// Memory-bound (AI ~4 flop/B); tensor (100.7MB) is L2-resident (192MB L2),
// so passes 2-4 never touch HBM. Per-tile input bbox is staged into LDS via
// CDNA5 async tensor path (global_load_async_to_lds_b128 + s_wait_asynccnt),
// bilinear taps served from LDS (ds_load_b32).

#define IMG 512
#define TILE 32
#define BW 64   // staged tile width  (floats) — worst pass (shear_x) needs <=61
#define BH 52   // staged tile height (rows)   — worst pass (shear_y) needs <=50

__global__ __launch_bounds__(256) void RandomAffine_warp_pass(
    const float* __restrict__ src, float* __restrict__ dst,
    float t00, float t01, float t02, float t10, float t11, float t12,
    int rollH, int rollW)
{
    __shared__ float tile[BH * BW];

    const int w0    = blockIdx.x * TILE;
    const int h0    = blockIdx.y * TILE;
    const int plane = blockIdx.z;                       // n*3 + c
    const size_t plane_off = (size_t)plane * (IMG * IMG);
    const int tid = threadIdx.x;

    // Same float chain as the reference (xs/ys -> gx/gy -> ix/iy).
    auto mapx = [&](float wf, float hf) -> float {
        float xs = (wf + 0.5f) * (2.0f / 512.0f) - 1.0f;
        float ys = (hf + 0.5f) * (2.0f / 512.0f) - 1.0f;
        float gx = t00 * xs + t01 * ys + t02;
        return ((gx + 1.0f) * 512.0f - 1.0f) * 0.5f;
    };
    auto mapy = [&](float wf, float hf) -> float {
        float xs = (wf + 0.5f) * (2.0f / 512.0f) - 1.0f;
        float ys = (hf + 0.5f) * (2.0f / 512.0f) - 1.0f;
        float gy = t10 * xs + t11 * ys + t12;
        return ((gy + 1.0f) * 512.0f - 1.0f) * 0.5f;
    };

    // ---- uniform (block-level) input bounding box from tile corners ----
    const float wlo = (float)w0, whi = (float)(w0 + TILE - 1);
    const float hlo = (float)h0, hhi = (float)(h0 + TILE - 1);
    float cx0 = mapx(wlo, hlo), cx1 = mapx(whi, hlo), cx2 = mapx(wlo, hhi), cx3 = mapx(whi, hhi);
    float cy0 = mapy(wlo, hlo), cy1 = mapy(whi, hlo), cy2 = mapy(wlo, hhi), cy3 = mapy(whi, hhi);
    float minix = fminf(fminf(cx0, cx1), fminf(cx2, cx3));
    float maxix = fmaxf(fmaxf(cx0, cx1), fmaxf(cx2, cx3));
    float miniy = fminf(fminf(cy0, cy1), fminf(cy2, cy3));
    float maxiy = fmaxf(fmaxf(cy0, cy1), fmaxf(cy2, cy3));
    int x0 = (int)floorf(minix);
    int x1 = (int)floorf(maxix) + 1;          // +1: right bilinear tap
    int y0 = (int)floorf(miniy);
    int y1 = (int)floorf(maxiy) + 1;
    int x0a = x0 & ~3;                        // 16B-align staging start (floor for negatives too)
    int bh  = y1 - y0 + 1;  if (bh > BH) bh = BH;  if (bh < 1) bh = 1;
    int bwq = ((x1 - x0a) >> 2) + 1;          // 16B chunks per row actually needed
    if (bwq > BW / 4) bwq = BW / 4;

    // ---- async stage bbox rows into LDS (CDNA5 async tensor path) ----
    const long total_f = (long)96 * IMG * IMG;             // whole-tensor clamp (no faults at edges)
    const uint32_t lds_base = (uint32_t)(uintptr_t)(&tile[0]);  // low 32b of generic ptr = LDS offset
    const int nchunk = (BW / 4) * bh;
    for (int cid = tid; cid < nchunk; cid += 256) {
        int r = cid >> 4;
        int q = cid & 15;
        if (q >= bwq) continue;                // columns beyond needed bbox: skip
        int yy = y0 + r;
        if (yy < 0 || yy > IMG - 1) continue;  // OOB rows never pass validity at gather time
        long el = (long)plane_off + (long)yy * IMG + (long)x0a + (long)(q * 4);
        if (el < 0) el = 0;
        if (el > total_f - 4) el = total_f - 4;
        const float* gp = src + el;
        uint32_t laddr = lds_base + (uint32_t)((r * BW + q * 4) * 4);
        asm volatile("global_load_async_to_lds_b128 %0, %1, off"
                     :: "v"(laddr), "v"(gp) : "memory");
    }
    asm volatile("s_wait_asynccnt 0" ::: "memory");
    __syncthreads();

    // ---- bilinear gather from LDS; one wave == one output row (coalesced) ----
    const int lw = tid & 31;
    const int lh = tid >> 5;
    const float fw = (float)(w0 + lw);
    #pragma unroll
    for (int k = 0; k < 4; ++k) {
        int h = h0 + lh + 8 * k;
        float fh = (float)h;
        float ix = mapx(fw, fh);
        float iy = mapy(fw, fh);
        float ix0f = floorf(ix), iy0f = floorf(iy);
        float wx1 = ix - ix0f, wy1 = iy - iy0f;
        float wx0 = 1.0f - wx1, wy0 = 1.0f - wy1;
        int ix0 = (int)ix0f, iy0 = (int)iy0f;
        int ix1 = ix0 + 1,  iy1 = iy0 + 1;
        bool vx0 = (ix0 >= 0) && (ix0 <= IMG - 1);
        bool vx1 = (ix1 >= 0) && (ix1 <= IMG - 1);
        bool vy0 = (iy0 >= 0) && (iy0 <= IMG - 1);
        bool vy1 = (iy1 >= 0) && (iy1 <= IMG - 1);
        int lx0 = min(max(ix0 - x0a, 0), BW - 1);
        int lx1 = min(max(ix1 - x0a, 0), BW - 1);
        int ly0 = min(max(iy0 - y0, 0), BH - 1);
        int ly1 = min(max(iy1 - y0, 0), BH - 1);
        // selects (not multiplies) so stale LDS NaNs can't leak via 0-weight
        float v00 = (vy0 && vx0) ? tile[ly0 * BW + lx0] : 0.0f;
        float v01 = (vy0 && vx1) ? tile[ly0 * BW + lx1] : 0.0f;
        float v10 = (vy1 && vx0) ? tile[ly1 * BW + lx0] : 0.0f;
        float v11 = (vy1 && vx1) ? tile[ly1 * BW + lx1] : 0.0f;
        float res = v00 * (wy0 * wx0) + v01 * (wy0 * wx1)
                  + v10 * (wy1 * wx0) + v11 * (wy1 * wx1);
        int oh = (h + rollH) & (IMG - 1);
        int ow = (w0 + lw + rollW) & (IMG - 1);
        dst[plane_off + (size_t)oh * IMG + ow] = res;
    }
}

extern "C" void kernel_launch(void* const* d_in, const int* in_sizes, int n_in,
                              void* d_out, int out_size, void* d_ws, size_t ws_size,
                              hipStream_t stream) {
    (void)in_sizes; (void)n_in; (void)out_size; (void)ws_size;
    const float* x  = (const float*)d_in[0];
    float* out = (float*)d_out;
    float* tmp = (float*)d_ws;   // needs >= 32*3*512*512*4 = 100,663,296 bytes

    // Affine matrices (same constants as reference _mats, computed in double, cast to f32)
    const double th = 10.0 * M_PI / 180.0;
    const float rc = (float)cos(th), rs = (float)sin(th);
    const float mx = (float)(1.0 / tan(0.9));
    const float my = (float)(1.0 / tan(1.1));

    dim3 grid(IMG / TILE, IMG / TILE, 96);
    dim3 block(256);

    // pass 1: rotation      x   -> tmp
    RandomAffine_warp_pass<<<grid, block, 0, stream>>>(x,   tmp, rc, -rs, 0.f,  rs,  rc, 0.f, 0, 0);
    // pass 2: scale         tmp -> out
    RandomAffine_warp_pass<<<grid, block, 0, stream>>>(tmp, out, 1.1f, 0.f, 0.f, 0.f, 1.1f, 0.f, 0, 0);
    // pass 3: shear_x       out -> tmp
    RandomAffine_warp_pass<<<grid, block, 0, stream>>>(out, tmp, 1.f,  mx, 0.f, 0.f, 1.f,  0.f, 0, 0);
    // pass 4: shear_y + fused rolls (8 on H, 12 on W)   tmp -> out
    RandomAffine_warp_pass<<<grid, block, 0, stream>>>(tmp, out, 1.f,  0.f, 0.f, my,  1.f,  0.f, 8, 12);
}